// GCN_13511967113874
// MI455X (gfx1250) — compile-verified
//
#include <hip/hip_runtime.h>

#define N_NODES 100000
#define IN_FT   256
#define OUT_FT  64

typedef __attribute__((ext_vector_type(2))) float v2f;
typedef __attribute__((ext_vector_type(4))) float v4f;
typedef __attribute__((ext_vector_type(8))) float v8f;

// ---------------------------------------------------------------------------
// GEMM: seq[N,256] x W[256,64] -> seq_fts[N,64] via V_WMMA_F32_16X16X4_F32.
// Each wave computes one 16(M) x 64(N) tile: 4 accumulators of v8f.
// W is staged to LDS pre-swizzled into per-lane B fragments:
//   B 4x16 layout: VGPR0 = {lanes0-15: K0, lanes16-31: K2}, VGPR1 = {K1, K3}
//   lds idx = ((ks*4 + nt)*32 + ln)*2 + j  holds  W[(ks*4 + 2*(ln>=16) + j)*64
//                                                   + nt*16 + (ln&15)]
// ---------------------------------------------------------------------------
__global__ __launch_bounds__(256) void gcn_gemm_wmma(
    const float* __restrict__ seq, const float* __restrict__ W,
    float* __restrict__ seq_fts, int n_mtiles) {
  __shared__ float lds_w[IN_FT * OUT_FT];  // 16384 floats = 64 KB (WGP: 320 KB)

  // Stage W with coalesced b128 loads, scatter into swizzled layout.
  // qword q covers W[krow, cq..cq+3]; for fixed krow the 4 consecutive cols
  // land at lds idx0, idx0+2, idx0+4, idx0+6.
  for (int q = (int)threadIdx.x; q < IN_FT * OUT_FT / 4; q += 256) {
    v4f w4 = *(const v4f*)(W + (size_t)q * 4);
    int krow = q >> 4;           // W row (K index)
    int cq   = (q & 15) << 2;    // starting column
    int ks = krow >> 2;
    int h  = (krow >> 1) & 1;    // lane-half select
    int j  = krow & 1;           // v2f element select
    int nt = cq >> 4;
    int idx0 = ks * 256 + nt * 64 + ((h << 4) + (cq & 15)) * 2 + j;
    lds_w[idx0 + 0] = w4.x;
    lds_w[idx0 + 2] = w4.y;
    lds_w[idx0 + 4] = w4.z;
    lds_w[idx0 + 6] = w4.w;
  }
  __syncthreads();

  const int wave = (int)(threadIdx.x >> 5);
  const int lane = (int)(threadIdx.x & 31);
  const int mtile = (int)blockIdx.x * 8 + wave;
  if (mtile >= n_mtiles) return;  // wave-uniform: EXEC stays all-1s for WMMA

  const int m0 = mtile * 16;
  // A fragment (16x4 f32): lanes 0-15 hold row (lane), K pair {k, k+1};
  // lanes 16-31 hold row (lane-16), K pair {k+2, k+3}.
  const float* arow =
      seq + (size_t)(m0 + (lane & 15)) * IN_FT + ((lane >> 4) << 1);

  v8f acc0 = {}, acc1 = {}, acc2 = {}, acc3 = {};
#pragma unroll 4
  for (int k = 0; k < IN_FT / 4; ++k) {
    v2f a = *(const v2f*)(arow + k * 4);
    __builtin_prefetch(arow + k * 4 + 64, 0, 3);  // WGP-scope prefetch ahead
    const v2f* bp = (const v2f*)&lds_w[k * 256];  // 4 N-tiles of B fragments
    acc0 = __builtin_amdgcn_wmma_f32_16x16x4_f32(false, a, false, bp[lane],
                                                 (short)0, acc0, false, false);
    acc1 = __builtin_amdgcn_wmma_f32_16x16x4_f32(false, a, false, bp[32 + lane],
                                                 (short)0, acc1, false, false);
    acc2 = __builtin_amdgcn_wmma_f32_16x16x4_f32(false, a, false, bp[64 + lane],
                                                 (short)0, acc2, false, false);
    acc3 = __builtin_amdgcn_wmma_f32_16x16x4_f32(false, a, false, bp[96 + lane],
                                                 (short)0, acc3, false, false);
  }

  // C/D 16x16 f32 layout: VGPR r <-> row r + 8*(lane>=16), col = lane&15
  float* orow =
      seq_fts + (size_t)(m0 + ((lane >> 4) << 3)) * OUT_FT + (lane & 15);
#pragma unroll
  for (int r = 0; r < 8; ++r) {
    orow[(size_t)r * OUT_FT + 0]  = acc0[r];
    orow[(size_t)r * OUT_FT + 16] = acc1[r];
    orow[(size_t)r * OUT_FT + 32] = acc2[r];
    orow[(size_t)r * OUT_FT + 48] = acc3[r];
  }
}

// ---------------------------------------------------------------------------
// SpMM scatter: one wave per edge, 2 features per lane.
// seq_fts (25.6 MB) and agg (25.6 MB) are L2-resident on the 192 MB L2,
// so gathers + global_atomic_add_f32 stay on-die; only the edge lists
// (19.2 MB) stream from HBM.
// ---------------------------------------------------------------------------
__global__ __launch_bounds__(256) void gcn_spmm_edges(
    const int* __restrict__ erow, const int* __restrict__ ecol,
    const float* __restrict__ evals, const float* __restrict__ x,
    float* __restrict__ agg, int n_edges) {
  int e = (int)blockIdx.x * 8 + (int)(threadIdx.x >> 5);
  if (e >= n_edges) return;
  int lane = (int)(threadIdx.x & 31);
  int r = erow[e];
  int c = ecol[e];
  float v = evals[e];
  v2f xv = ((const v2f*)(x + (size_t)c * OUT_FT))[lane];
  float* dst = agg + (size_t)r * OUT_FT + lane * 2;
  atomicAdd(dst, v * xv.x);
  atomicAdd(dst + 1, v * xv.y);
}

__global__ __launch_bounds__(256) void gcn_zero4(v4f* __restrict__ p, int n4) {
  int i = (int)blockIdx.x * 256 + (int)threadIdx.x;
  if (i < n4) p[i] = (v4f){0.0f, 0.0f, 0.0f, 0.0f};
}

// out[i] = relu(out[i] + b[i % 64]), vectorized x4 (64 is a multiple of 4).
__global__ __launch_bounds__(256) void gcn_bias_relu4(
    v4f* __restrict__ out, const v4f* __restrict__ b4, int n4) {
  int i = (int)blockIdx.x * 256 + (int)threadIdx.x;
  if (i < n4) {
    v4f t = out[i];
    v4f bb = b4[i & (OUT_FT / 4 - 1)];
    t += bb;
    t.x = t.x > 0.0f ? t.x : 0.0f;
    t.y = t.y > 0.0f ? t.y : 0.0f;
    t.z = t.z > 0.0f ? t.z : 0.0f;
    t.w = t.w > 0.0f ? t.w : 0.0f;
    out[i] = t;
  }
}

extern "C" void kernel_launch(void* const* d_in, const int* in_sizes, int n_in,
                              void* d_out, int out_size, void* d_ws,
                              size_t ws_size, hipStream_t stream) {
  const float* seq   = (const float*)d_in[0];  // [1, N, 256] fp32
  const int*   erow  = (const int*)d_in[1];    // [E] int32
  const int*   ecol  = (const int*)d_in[2];    // [E] int32
  const float* evals = (const float*)d_in[3];  // [E] fp32
  const float* W     = (const float*)d_in[4];  // [256, 64] fp32
  const float* b     = (const float*)d_in[5];  // [64] fp32

  const int n_nodes = in_sizes[0] / IN_FT;  // 100000
  const int n_edges = in_sizes[1];          // 1600000
  const int n_feat  = n_nodes * OUT_FT;
  const int n4      = n_feat / 4;

  float* out     = (float*)d_out;        // first output: relu(agg + b)
  float* seq_fts = out + (size_t)n_feat; // second output: seq @ W

  // 1) zero the aggregation target (we atomically accumulate into d_out)
  gcn_zero4<<<(n4 + 255) / 256, 256, 0, stream>>>((v4f*)out, n4);

  // 2) fp32 WMMA GEMM -> seq_fts (second half of d_out)
  const int n_mtiles = n_nodes / 16;  // 6250, exact
  gcn_gemm_wmma<<<(n_mtiles + 7) / 8, 256, 0, stream>>>(seq, W, seq_fts,
                                                        n_mtiles);

  // 3) edge scatter with fp32 atomics (L2-resident working set)
  gcn_spmm_edges<<<(n_edges + 7) / 8, 256, 0, stream>>>(erow, ecol, evals,
                                                        seq_fts, out, n_edges);

  // 4) in-place bias + ReLU on the aggregated output
  gcn_bias_relu4<<<(n4 + 255) / 256, 256, 0, stream>>>((v4f*)out,
                                                       (const v4f*)b, n4);
}